// DetectionLoss_9251359556112
// MI455X (gfx1250) — compile-verified
//
#include <hip/hip_runtime.h>
#include <hip/hip_bf16.h>

// Problem constants from the reference
#define BATCH 16
#define GBOX  32
#define NANCH 16384
#define NCLS  80

typedef float v2f __attribute__((ext_vector_type(2)));
typedef float v8f __attribute__((ext_vector_type(8)));

// Full wave32 sum using V_WMMA_F32_16X16X4_F32.
// A = ones(16x4), B VGPR0 = lane value (K0/K2 slots), VGPR1 = 0 (K1/K3 slots).
// => D[m][n] = v[lane n] + v[lane n+16] ("column sums"), identical for all m,
// and every lane's c[0] holds colsum[lane & 15]. Four xor-folds over the
// 16-wide pattern complete the reduction; result broadcast to all lanes.
// Requires all 32 lanes active (call unconditionally from full waves).
__device__ __forceinline__ float wave32_sum_wmma(float v) {
    v2f a;  a.x = 1.0f; a.y = 1.0f;
    v2f bm; bm.x = v;   bm.y = 0.0f;
    v8f c = {};
    c = __builtin_amdgcn_wmma_f32_16x16x4_f32(
            /*neg_a=*/false, a, /*neg_b=*/false, bm,
            /*c_mod=*/(short)0, c, /*reuse_a=*/false, /*reuse_b=*/false);
    float s = c[0];
    s += __shfl_xor(s, 1);
    s += __shfl_xor(s, 2);
    s += __shfl_xor(s, 4);
    s += __shfl_xor(s, 8);
    return s;
}

// ---------------------------------------------------------------------------
// Kernel 1: per-(b,g) IoU argmax over N anchors. 512 blocks x 256 threads.
// jnp.argmax tie-break = first (lowest) index: strict '>' within a thread's
// increasing-index stride; lower index wins on equal value in the tree.
// ---------------------------------------------------------------------------
__global__ __launch_bounds__(256)
void match_kernel(const float* __restrict__ box_preds,
                  const float* __restrict__ gt_boxes,
                  int* __restrict__ matched) {
    const int bg = blockIdx.x;
    const int b = bg >> 5;        // / GBOX
    const int tid = threadIdx.x;

    const float4 gt = ((const float4*)gt_boxes)[bg];
    const float garea = (gt.z - gt.x) * (gt.w - gt.y);
    const float4* __restrict__ boxes = ((const float4*)box_preds) + (size_t)b * NANCH;

    float best = -1.0f;
    int   bidx = 0;
    for (int i = tid; i < NANCH; i += 256) {
        if (i + 256 < NANCH) __builtin_prefetch(boxes + i + 256, 0, 0);
        const float4 p = boxes[i];
        const float parea = (p.z - p.x) * (p.w - p.y);
        const float lx = fmaxf(gt.x, p.x);
        const float ly = fmaxf(gt.y, p.y);
        const float rx = fminf(gt.z, p.z);
        const float ry = fminf(gt.w, p.w);
        const float w = fmaxf(rx - lx, 0.0f);
        const float h = fmaxf(ry - ly, 0.0f);
        const float inter = w * h;
        const float iou = inter / (garea + parea - inter);
        if (iou > best) { best = iou; bidx = i; }
    }

    __shared__ float sval[256];
    __shared__ int   sidx[256];
    sval[tid] = best; sidx[tid] = bidx;
    __syncthreads();
    for (int s = 128; s > 0; s >>= 1) {
        if (tid < s) {
            const float ov = sval[tid + s];
            const int   oi = sidx[tid + s];
            if (ov > sval[tid] || (ov == sval[tid] && oi < sidx[tid])) {
                sval[tid] = ov; sidx[tid] = oi;
            }
        }
        __syncthreads();
    }
    if (tid == 0) matched[bg] = sidx[0];
}

// ---------------------------------------------------------------------------
// Kernel 2: per-(b,g) loss partials. 512 blocks x 128 threads (4 full waves).
// BCE over 80 gathered logits + SmoothL1 over 4 coords; WMMA wave reductions.
// ---------------------------------------------------------------------------
__global__ __launch_bounds__(128)
void loss_kernel(const float* __restrict__ cls_preds,
                 const float* __restrict__ box_preds,
                 const float* __restrict__ gt_boxes,
                 const int*   __restrict__ gt_labels,
                 const int*   __restrict__ matched,
                 float* __restrict__ pcls,
                 float* __restrict__ pbox) {
    const int bg  = blockIdx.x;
    const int b   = bg >> 5;
    const int tid = threadIdx.x;
    const int m   = matched[bg];

    const float* __restrict__ x = cls_preds + ((size_t)b * NANCH + m) * NCLS;
    int label = gt_labels[bg];
    label = (label < 0) ? 0 : (label > NCLS - 1 ? NCLS - 1 : label);

    float cl = 0.0f;
    for (int c = tid; c < NCLS; c += 128) {
        const float xv = x[c];
        const float t  = (c == label) ? 1.0f : 0.0f;
        cl += fmaxf(xv, 0.0f) - xv * t + log1pf(expf(-fabsf(xv)));
    }

    float bl = 0.0f;
    if (tid < 4) {
        const float mv = box_preds[((size_t)b * NANCH + m) * 4 + tid];
        const float gv = gt_boxes[(size_t)bg * 4 + tid];
        const float d  = mv - gv;
        const float ad = fabsf(d);
        bl = (ad < 1.0f) ? 0.5f * d * d : ad - 0.5f;
    }

    // All 128 threads (4 full waves) execute the WMMA reductions.
    const float cw = wave32_sum_wmma(cl);
    const float bw = wave32_sum_wmma(bl);

    __shared__ float wcls[4], wbox[4];
    const int wave = tid >> 5;
    if ((tid & 31) == 0) { wcls[wave] = cw; wbox[wave] = bw; }
    __syncthreads();
    if (tid == 0) {
        pcls[bg] = wcls[0] + wcls[1] + wcls[2] + wcls[3];
        pbox[bg] = wbox[0] + wbox[1] + wbox[2] + wbox[3];
    }
}

// ---------------------------------------------------------------------------
// Kernel 3: deterministic finalize. 1 block x 256 threads (8 full waves).
// ---------------------------------------------------------------------------
__global__ __launch_bounds__(256)
void finalize_kernel(const float* __restrict__ pcls,
                     const float* __restrict__ pbox,
                     float* __restrict__ out) {
    const int tid = threadIdx.x;
    float c = 0.0f, bx = 0.0f;
    for (int i = tid; i < BATCH * GBOX; i += 256) {
        c  += pcls[i];
        bx += pbox[i];
    }
    const float cw = wave32_sum_wmma(c);
    const float bw = wave32_sum_wmma(bx);

    __shared__ float sc[8], sb[8];
    const int wave = tid >> 5;
    if ((tid & 31) == 0) { sc[wave] = cw; sb[wave] = bw; }
    __syncthreads();
    if (tid == 0) {
        float tc = 0.0f, tb = 0.0f;
        #pragma unroll
        for (int w = 0; w < 8; ++w) { tc += sc[w]; tb += sb[w]; }
        out[0] = tc * (1.0f / (float)(BATCH * GBOX * NCLS))
               + tb * (1.0f / (float)(BATCH * GBOX * 4));
    }
}

extern "C" void kernel_launch(void* const* d_in, const int* in_sizes, int n_in,
                              void* d_out, int out_size, void* d_ws, size_t ws_size,
                              hipStream_t stream) {
    const float* cls_preds = (const float*)d_in[0];   // [B,N,C] f32
    const float* box_preds = (const float*)d_in[1];   // [B,N,4] f32
    const float* gt_boxes  = (const float*)d_in[2];   // [B,G,4] f32
    const int*   gt_labels = (const int*)  d_in[3];   // [B,G] int
    float* out = (float*)d_out;                       // scalar

    char* ws = (char*)d_ws;
    int*   matched = (int*)  (ws + 0);      // 512 ints
    float* pcls    = (float*)(ws + 2048);   // 512 floats
    float* pbox    = (float*)(ws + 4096);   // 512 floats

    match_kernel<<<BATCH * GBOX, 256, 0, stream>>>(box_preds, gt_boxes, matched);
    loss_kernel<<<BATCH * GBOX, 128, 0, stream>>>(cls_preds, box_preds, gt_boxes,
                                                  gt_labels, matched, pcls, pbox);
    finalize_kernel<<<1, 256, 0, stream>>>(pcls, pbox, out);
}